// Node_GNN_15401752723891
// MI455X (gfx1250) — compile-verified
//
#include <hip/hip_runtime.h>
#include <math.h>

// ---------------------------------------------------------------------------
// Node GNN (3-layer GCN over 8 independent feature columns) for MI455X/gfx1250
//   N = 50000 nodes, E = 1.6M edges, F = 8 cols, D = 64 hidden
// Layer-1 GEMMs via V_WMMA_F32_16X16X4_F32 (wave32, f32 precision).
// ---------------------------------------------------------------------------

#define NNODES 50000
#define NEDGES 1600000
#define NFEATS 8
#define DHID   64
#define NROWS  (NNODES * NFEATS)   // 400000 "rows" = (node, col) pairs
#define NPAD   50048               // aligned size for dinv buffer

typedef float v2f __attribute__((ext_vector_type(2)));
typedef float v8f __attribute__((ext_vector_type(8)));

// Branch-free ELU: v_exp_f32 (TRANS, co-executes with WMMA) + v_cndmask.
__device__ __forceinline__ float elu1(float x) {
    float en = __expf(x) - 1.0f;
    return x > 0.0f ? x : en;
}

// ---------------------------------------------------------------- zero fill
__global__ void k_zero(float* __restrict__ p, long n) {
    long i = (long)blockIdx.x * blockDim.x + threadIdx.x;
    if (i < n) p[i] = 0.0f;
}

// ------------------------------------------------------------- degree count
__global__ void k_degree(const long long* __restrict__ ei, float* __restrict__ deg) {
    int e = blockIdx.x * blockDim.x + threadIdx.x;
    if (e < NEDGES) {
        int c = (int)ei[NEDGES + e];
        atomicAdd(&deg[c], 1.0f);
    }
}

// --------------------------------------------------------- deg -> rsqrt(deg)
__global__ void k_dinv(float* __restrict__ deg) {
    int n = blockIdx.x * blockDim.x + threadIdx.x;
    if (n < NNODES) {
        float d = deg[n];
        deg[n] = d > 0.0f ? rsqrtf(d) : 0.0f;
    }
}

// ---------------- edge table: (row, col, norm) as int32/int32/f32 per edge
__global__ void k_edgeprep(const long long* __restrict__ ei,
                           const float* __restrict__ dinv,
                           int* __restrict__ eri, int* __restrict__ eci,
                           float* __restrict__ enw) {
    int e = blockIdx.x * blockDim.x + threadIdx.x;
    if (e >= NEDGES) return;
    int r = (int)ei[e];
    int c = (int)ei[NEDGES + e];
    eri[e] = r;
    eci[e] = c;
    enw[e] = dinv[r] * dinv[c];
}

// ------------------------- layer 0 scatter: s[c, 0..7] += norm * X[r, 0..7]
__global__ void k_l0_edge(const int* __restrict__ eri, const int* __restrict__ eci,
                          const float* __restrict__ enw,
                          const float* __restrict__ X,
                          float* __restrict__ s) {
    int e = blockIdx.x * blockDim.x + threadIdx.x;
    if (e >= NEDGES) return;
    int r = eri[e], c = eci[e];
    float nrm = enw[e];
    const float4* xr = (const float4*)(X + (long)r * NFEATS);
    float4 x0 = xr[0], x1 = xr[1];
    float* sc = s + (long)c * NFEATS;
    atomicAdd(sc + 0, nrm * x0.x);
    atomicAdd(sc + 1, nrm * x0.y);
    atomicAdd(sc + 2, nrm * x0.z);
    atomicAdd(sc + 3, nrm * x0.w);
    atomicAdd(sc + 4, nrm * x1.x);
    atomicAdd(sc + 5, nrm * x1.y);
    atomicAdd(sc + 6, nrm * x1.z);
    atomicAdd(sc + 7, nrm * x1.w);
}

// ---------------------------------------------------------------------------
// Layer-1 GEMM with WMMA f32 16x16x4.
// Row r = n*8 + c: h0[r,k] = elu(X[r]*H00[k] + s[r]*H01[k]) generated in
// registers; outputs  h1p[r,:] = h0 @ H1_0   and   y[r,:] = h0 @ H1_1.
// One wave computes a 16-row x 64-col tile for BOTH weight matrices.
// Weights pre-packed in LDS as row-pair-interleaved v2f so each B fragment
// is a single aligned ds_load_b64; H0 taps packed as float4 (one b128/step).
// ---------------------------------------------------------------------------
__global__ __launch_bounds__(256)
void k_l1_gemm(const float* __restrict__ Xf,   // [NROWS]  (= X flattened)
               const float* __restrict__ s,    // [NROWS]
               const float* __restrict__ H00,  // [64]
               const float* __restrict__ H01,  // [64]
               const float* __restrict__ W0,   // H1_0 [64][64] row-major (k,n)
               const float* __restrict__ W1,   // H1_1 [64][64]
               float* __restrict__ h1p,        // [NROWS][64]
               float* __restrict__ y)          // [NROWS][64]
{
    // sWxp[p*64 + n] = { W[2p][n], W[2p+1][n] }  for row-pair p = 0..31
    __shared__ v2f sW0p[32 * 64];
    __shared__ v2f sW1p[32 * 64];
    // sHp[p] = { H00[2p], H01[2p], H00[2p+1], H01[2p+1] }
    __shared__ float4 sHp[32];

    for (int i = threadIdx.x; i < 32 * 64; i += 256) {
        int p = i >> 6, n = i & 63;
        v2f w0, w1;
        w0.x = W0[(2 * p) * 64 + n];
        w0.y = W0[(2 * p + 1) * 64 + n];
        w1.x = W1[(2 * p) * 64 + n];
        w1.y = W1[(2 * p + 1) * 64 + n];
        sW0p[i] = w0;
        sW1p[i] = w1;
    }
    if (threadIdx.x < 32) {
        int k0 = threadIdx.x * 2;
        float4 h;
        h.x = H00[k0];
        h.y = H01[k0];
        h.z = H00[k0 + 1];
        h.w = H01[k0 + 1];
        sHp[threadIdx.x] = h;
    }
    __syncthreads();

    const int wave = threadIdx.x >> 5;
    const int lane = threadIdx.x & 31;
    const int half = lane >> 4;     // 0: lanes 0-15, 1: lanes 16-31
    const int l16  = lane & 15;
    const long R0 = ((long)blockIdx.x * 8 + wave) * 16;   // 25000 tiles total

    // A-fragment row scalars (M = lane&15 for both lane halves)
    const long rM = R0 + l16;
    const float xv = Xf[rM];
    const float sv = s[rM];

    v8f acc0[4] = {};   // h0 @ H1_0, four 16-wide N tiles
    v8f acc1[4] = {};   // h0 @ H1_1

    #pragma unroll
    for (int t = 0; t < 16; ++t) {          // K = 4*t .. 4*t+3
        const int p = 2 * t + half;         // row-pair index: k0 = 2p, k1 = 2p+1
        const float4 hc = sHp[p];
        v2f a;
        a.x = elu1(xv * hc.x + sv * hc.y);
        a.y = elu1(xv * hc.z + sv * hc.w);

        #pragma unroll
        for (int nt = 0; nt < 4; ++nt) {
            const int n = nt * 16 + l16;
            v2f b0 = sW0p[p * 64 + n];
            v2f b1 = sW1p[p * 64 + n];
            acc0[nt] = __builtin_amdgcn_wmma_f32_16x16x4_f32(
                false, a, false, b0, (short)0, acc0[nt], false, false);
            acc1[nt] = __builtin_amdgcn_wmma_f32_16x16x4_f32(
                false, a, false, b1, (short)0, acc1[nt], false, false);
        }
    }

    // C/D layout: VGPR i -> (lanes 0-15: M=i, lanes 16-31: M=8+i), N = lane&15
    #pragma unroll
    for (int nt = 0; nt < 4; ++nt) {
        #pragma unroll
        for (int i = 0; i < 8; ++i) {
            const long rr = R0 + half * 8 + i;
            const long idx = rr * 64 + nt * 16 + l16;
            h1p[idx] = acc0[nt][i];
            y[idx]   = acc1[nt][i];
        }
    }
}

// ------------- layer 1 message pass: h1p[col,c,d] += norm * y[row,c,d]
// 128 threads per edge; each thread moves one float4 (c = q/16, d4 = (q%16)*4)
__global__ __launch_bounds__(256)
void k_l1_edge(const int* __restrict__ eri, const int* __restrict__ eci,
               const float* __restrict__ enw,
               const float* __restrict__ y,
               float* __restrict__ h1p) {
    const long t = (long)blockIdx.x * 256 + threadIdx.x;   // exactly E*128 threads
    const int e  = (int)(t >> 7);
    const int q  = (int)(t & 127);
    const int c  = q >> 4;
    const int d4 = (q & 15) * 4;

    const int r   = eri[e];
    const int cn  = eci[e];
    const float nrm = enw[e];

    const float4 v = *(const float4*)(y + (((long)r * NFEATS + c) << 6) + d4);
    float* dst = h1p + (((long)cn * NFEATS + c) << 6) + d4;
    atomicAdd(dst + 0, nrm * v.x);
    atomicAdd(dst + 1, nrm * v.y);
    atomicAdd(dst + 2, nrm * v.z);
    atomicAdd(dst + 3, nrm * v.w);
}

// ------------------------------------------------- in-place ELU (float4)
__global__ void k_elu4(float4* __restrict__ p, long n4) {
    long i = (long)blockIdx.x * blockDim.x + threadIdx.x;
    if (i >= n4) return;
    float4 v = p[i];
    v.x = elu1(v.x);
    v.y = elu1(v.y);
    v.z = elu1(v.z);
    v.w = elu1(v.w);
    p[i] = v;
}

// ------------- layer 2 dual GEMV: out[r] = h1[r,:]@H2_0 ; b2[r] = h1@H2_1
__global__ __launch_bounds__(256)
void k_l2_row(const float* __restrict__ h1,
              const float* __restrict__ H20,   // [64]
              const float* __restrict__ H21,   // [64]
              float* __restrict__ out,         // [NROWS] == d_out [N][8]
              float* __restrict__ b2) {        // [NROWS]
    __shared__ float sA[64], sB[64];
    if (threadIdx.x < 64) {
        sA[threadIdx.x] = H20[threadIdx.x];
        sB[threadIdx.x] = H21[threadIdx.x];
    }
    __syncthreads();
    const long r = (long)blockIdx.x * 256 + threadIdx.x;
    if (r >= NROWS) return;
    const float4* hp = (const float4*)(h1 + r * 64);
    float a = 0.0f, b = 0.0f;
    #pragma unroll
    for (int i = 0; i < 16; ++i) {
        float4 h = hp[i];
        a += h.x * sA[4 * i] + h.y * sA[4 * i + 1] + h.z * sA[4 * i + 2] + h.w * sA[4 * i + 3];
        b += h.x * sB[4 * i] + h.y * sB[4 * i + 1] + h.z * sB[4 * i + 2] + h.w * sB[4 * i + 3];
    }
    out[r] = a;
    b2[r]  = b;
}

// ------------- layer 2 scatter: out[col, 0..7] += norm * b2[row, 0..7]
__global__ void k_l2_edge(const int* __restrict__ eri, const int* __restrict__ eci,
                          const float* __restrict__ enw,
                          const float* __restrict__ b2,
                          float* __restrict__ out) {
    int e = blockIdx.x * blockDim.x + threadIdx.x;
    if (e >= NEDGES) return;
    int r = eri[e], c = eci[e];
    float nrm = enw[e];
    const float4* br = (const float4*)(b2 + (long)r * NFEATS);
    float4 v0 = br[0], v1 = br[1];
    float* oc = out + (long)c * NFEATS;
    atomicAdd(oc + 0, nrm * v0.x);
    atomicAdd(oc + 1, nrm * v0.y);
    atomicAdd(oc + 2, nrm * v0.z);
    atomicAdd(oc + 3, nrm * v0.w);
    atomicAdd(oc + 4, nrm * v1.x);
    atomicAdd(oc + 5, nrm * v1.y);
    atomicAdd(oc + 6, nrm * v1.z);
    atomicAdd(oc + 7, nrm * v1.w);
}

// ---------------------------------------------------------------------------
extern "C" void kernel_launch(void* const* d_in, const int* in_sizes, int n_in,
                              void* d_out, int out_size, void* d_ws, size_t ws_size,
                              hipStream_t stream) {
    const float*     X   = (const float*)d_in[0];     // [N][8]
    const long long* ei  = (const long long*)d_in[1]; // [2][E]
    const float*     H00 = (const float*)d_in[2];     // [64]
    const float*     H01 = (const float*)d_in[3];     // [64]
    const float*     W0  = (const float*)d_in[4];     // [64][64]
    const float*     W1  = (const float*)d_in[5];     // [64][64]
    const float*     H20 = (const float*)d_in[6];     // [64]
    const float*     H21 = (const float*)d_in[7];     // [64]
    float* out = (float*)d_out;                       // [N][8] flat = [NROWS]

    // Workspace layout (4-byte units):
    float* ws   = (float*)d_ws;
    float* dinv = ws;                        // NPAD      (deg, then rsqrt(deg))
    float* s    = dinv + NPAD;               // NROWS     (layer-0 scalar agg)
    float* b2   = s + NROWS;                 // NROWS     (layer-2 Hb GEMV)
    int*   eri  = (int*)(b2 + NROWS);        // NEDGES    (edge row)
    int*   eci  = eri + NEDGES;              // NEDGES    (edge col)
    float* enw  = (float*)(eci + NEDGES);    // NEDGES    (edge norm)
    float* y    = enw + NEDGES;              // NROWS*64  (h0 @ H1_1)
    float* h1p  = y + (long)NROWS * 64;      // NROWS*64  (h0 @ H1_0 + agg -> h1)

    const int TB = 256;

    // 1) zero deg + s
    {
        long ztot = (long)NPAD + NROWS;
        k_zero<<<(unsigned)((ztot + TB - 1) / TB), TB, 0, stream>>>(dinv, ztot);
    }
    // 2) degree + rsqrt
    k_degree<<<(NEDGES + TB - 1) / TB, TB, 0, stream>>>(ei, dinv);
    k_dinv<<<(NNODES + TB - 1) / TB, TB, 0, stream>>>(dinv);
    // 3) edge table (row, col, norm)
    k_edgeprep<<<(NEDGES + TB - 1) / TB, TB, 0, stream>>>(ei, dinv, eri, eci, enw);
    // 4) layer-0 scalar scatter
    k_l0_edge<<<(NEDGES + TB - 1) / TB, TB, 0, stream>>>(eri, eci, enw, X, s);
    // 5) layer-1 GEMMs via WMMA (25000 row-tiles, 8 waves/block)
    k_l1_gemm<<<NROWS / (16 * 8), TB, 0, stream>>>(X, s, H00, H01, W0, W1, h1p, y);
    // 6) layer-1 message pass (E * 128 threads)
    {
        long total = (long)NEDGES * 128;
        k_l1_edge<<<(unsigned)(total / TB), TB, 0, stream>>>(eri, eci, enw, y, h1p);
    }
    // 7) ELU in place -> h1 (float4)
    {
        long n4 = (long)NROWS * 64 / 4;
        k_elu4<<<(unsigned)((n4 + TB - 1) / TB), TB, 0, stream>>>((float4*)h1p, n4);
    }
    // 8) layer-2 dual GEMV (writes a-part into d_out, b-part into b2)
    k_l2_row<<<(NROWS + TB - 1) / TB, TB, 0, stream>>>(h1p, H20, H21, out, b2);
    // 9) layer-2 scalar scatter onto d_out
    k_l2_edge<<<(NEDGES + TB - 1) / TB, TB, 0, stream>>>(eri, eci, enw, b2, out);
}